// CA_23175643529789
// MI455X (gfx1250) — compile-verified
//
#include <hip/hip_runtime.h>

// ---------------------------------------------------------------------------
// NCA-style stack: depthwise3x3(dm=3) -> [1x1 conv + BN] x 11 -> 1x1 -> +x
// B=16, H=W=1024, C=1.  Memory-bound (134 MB @ 23.3 TB/s ~ 5.8us floor).
// Strategy:
//   kernel 1 (1 thread): fold all convs/BNs into {K[9],Kb, A[10],C[10], Af,Cf}
//   kernel 2: per block, TDM-load 18 full image rows (72KB) into LDS with one
//             tensor_load_to_lds descriptor, then 9-tap stencil + 10x(fma,max)
//             chain per pixel, residual from the center tap, b128 stores.
// ---------------------------------------------------------------------------

typedef __attribute__((ext_vector_type(4))) unsigned int v4u;
typedef __attribute__((ext_vector_type(8))) int          v8i;
typedef __attribute__((ext_vector_type(4))) int          v4i;

#define CA_W   1024
#define CA_H   1024
#define CA_TH  16            // output rows per block (LDS tile = TH+2 rows)

// ---------------- parameter folding (tiny, 1 thread) -----------------------
__global__ void ca_fold(const float* __restrict__ dwk,   // [3,3,1,3] -> [9][3]
                        const float* __restrict__ dwb,   // [3]
                        const float* __restrict__ w0,    // [3]
                        const float* __restrict__ b0,    // [1]
                        const float* __restrict__ ws,    // [10]
                        const float* __restrict__ bs,    // [10]
                        const float* __restrict__ gamma, // [11]
                        const float* __restrict__ beta,  // [11]
                        const float* __restrict__ mmean, // [11]
                        const float* __restrict__ mvar,  // [11]
                        const float* __restrict__ wf,    // [1]
                        const float* __restrict__ bf,    // [1]
                        float* __restrict__ P)           // [32] folded consts
{
    if (threadIdx.x != 0 || blockIdx.x != 0) return;
    float s[11], t[11];
    for (int i = 0; i < 11; ++i) {
        float sv = gamma[i] * rsqrtf(mvar[i] + 1e-3f);
        s[i] = sv;
        t[i] = beta[i] - mmean[i] * sv;
    }
    // fold depthwise(3ch) + first 1x1 (3->1) into one 9-tap kernel
    for (int k = 0; k < 9; ++k) {
        float acc = 0.f;
        for (int c = 0; c < 3; ++c) acc += w0[c] * dwk[k * 3 + c];
        P[k] = acc;
    }
    float kb = b0[0];
    for (int c = 0; c < 3; ++c) kb += w0[c] * dwb[c];
    P[9] = kb;
    // fold BN_j into middle conv j:  r_{j+1} = relu(r_j*A[j] + C[j])
    for (int j = 0; j < 10; ++j) {
        P[10 + j] = s[j] * ws[j];
        P[20 + j] = t[j] * ws[j] + bs[j];
    }
    // fold BN_10 into final conv:  dx = r_10*Af + Cf
    P[30] = s[10] * wf[0];
    P[31] = t[10] * wf[0] + bf[0];
}

// ---------------- main fused kernel ----------------------------------------
__global__ __launch_bounds__(256) void ca_main(const float* __restrict__ x,
                                               float* __restrict__ out,
                                               const float* __restrict__ P)
{
    extern __shared__ float tile[];                 // (CA_TH+2) * CA_W floats
    const int t  = threadIdx.x;                     // 0..255
    const int y0 = blockIdx.x * CA_TH;
    const int b  = blockIdx.y;

    // folded parameters (uniform addresses -> scalar loads)
    float K[9];
#pragma unroll
    for (int k = 0; k < 9; ++k) K[k] = P[k];
    const float Kb = P[9];
    float A[10], Cc[10];
#pragma unroll
    for (int j = 0; j < 10; ++j) { A[j] = P[10 + j]; Cc[j] = P[20 + j]; }
    const float Af = P[30], Cf = P[31];

    // SAME padding: zero top/bottom halo rows at the image border (DS stores)
    const float4 z4 = make_float4(0.f, 0.f, 0.f, 0.f);
    if (y0 == 0)
        reinterpret_cast<float4*>(tile)[t] = z4;                       // row 0
    if (y0 + CA_TH == CA_H)
        reinterpret_cast<float4*>(tile + (CA_TH + 1) * CA_W)[t] = z4;  // row TH+1

    // One TDM descriptor per block: 17-18 contiguous image rows -> LDS.
    // Issued by wave 0 only (TDM is a wave-level op, EXEC ignored).
    if (t < 32) {
        const int rlo = (y0 == 0) ? 0 : (y0 - 1);
        const int rhi = (y0 + CA_TH + 1 > CA_H) ? CA_H : (y0 + CA_TH + 1);
        const unsigned n = (unsigned)((rhi - rlo) * CA_W);   // elements <= 18432
        const unsigned long long ga =
            (unsigned long long)(uintptr_t)(x + ((size_t)b * CA_H + (size_t)rlo) * CA_W);
        const unsigned lds_addr =
            (unsigned)(unsigned long long)(uintptr_t)(tile + (rlo - (y0 - 1)) * CA_W);

        // D# group0: count=1 | lds_addr | global_addr[56:0] | type=2
        v4u g0 = { 1u,
                   lds_addr,
                   (unsigned)(ga & 0xFFFFFFFFull),
                   (unsigned)((ga >> 32) & 0x1FFFFFFull) | (2u << 30) };
        // D# group1: data_size=4B; tensor_dim0=n; tensor_dim1=1; 1-D tile:
        // tile_dim0=n (fits 16 bits), tile_dim1/2 unused; stride0=n.
        v8i g1 = { (int)0x20000u,                       // mask=0, data_size=2(4B)
                   (int)((n & 0xFFFFu) << 16),          // tensor_dim0[15:0]
                   (int)((n >> 16) | (1u << 16)),       // tensor_dim0[31:16], dim1=1
                   (int)((n & 0xFFFFu) << 16),          // tile_dim0 = n
                   0,                                   // tile_dim1=tile_dim2=0
                   (int)n,                              // tensor_dim0_stride[31:0]
                   0, 0 };
        v4i g2 = { 0, 0, 0, 0 };
        v4i g3 = { 0, 0, 0, 0 };
#if defined(__clang_major__) && (__clang_major__ >= 23)
        v8i g4 = { 0, 0, 0, 0, 0, 0, 0, 0 };
        __builtin_amdgcn_tensor_load_to_lds(g0, g1, g2, g3, g4, 0);
#else
        __builtin_amdgcn_tensor_load_to_lds(g0, g1, g2, g3, 0);
#endif
        __builtin_amdgcn_s_wait_tensorcnt(0);
    }
    __syncthreads();

    // ---- compute: each thread owns 4 consecutive columns, slides a 3-row
    //      window down CA_TH rows.  b128 LDS loads, 2 edge scalars per row.
    const int c0 = t << 2;
    float  L[3], R[3];
    float4 M[3];

    auto loadRow = [&](int slot, int rr) {
        const float* p = tile + rr * CA_W + c0;
        M[slot] = *reinterpret_cast<const float4*>(p);   // ds_load_b128 (16B aligned)
        float lv = p[-1];
        float rv = p[4];
        L[slot] = (c0 == 0)           ? 0.f : lv;        // SAME pad, left column
        R[slot] = (c0 + 4 >= CA_W)    ? 0.f : rv;        // SAME pad, right column
    };

    loadRow(0, 0);
    loadRow(1, 1);

#pragma unroll
    for (int rr = 0; rr < CA_TH; ++rr) {
        loadRow((rr + 2) % 3, rr + 2);

        const int i0 = rr % 3, i1 = (rr + 1) % 3, i2 = (rr + 2) % 3;
        float w0r[6] = { L[i0], M[i0].x, M[i0].y, M[i0].z, M[i0].w, R[i0] };
        float w1r[6] = { L[i1], M[i1].x, M[i1].y, M[i1].z, M[i1].w, R[i1] };
        float w2r[6] = { L[i2], M[i2].x, M[i2].y, M[i2].z, M[i2].w, R[i2] };

        float o[4];
#pragma unroll
        for (int j = 0; j < 4; ++j) {
            float pre = Kb;
#pragma unroll
            for (int d = 0; d < 3; ++d) {
                pre = fmaf(K[0 * 3 + d], w0r[j + d], pre);
                pre = fmaf(K[1 * 3 + d], w1r[j + d], pre);
                pre = fmaf(K[2 * 3 + d], w2r[j + d], pre);
            }
            float r = fmaxf(pre, 0.f);
#pragma unroll
            for (int i = 0; i < 10; ++i)
                r = fmaxf(fmaf(r, A[i], Cc[i]), 0.f);
            o[j] = w1r[j + 1] + fmaf(r, Af, Cf);         // residual = center tap
        }
        float* dst = out + ((size_t)(b * CA_H + (y0 + rr))) * CA_W + c0;
        *reinterpret_cast<float4*>(dst) = make_float4(o[0], o[1], o[2], o[3]);
    }
}

// ---------------------------------------------------------------------------
extern "C" void kernel_launch(void* const* d_in, const int* in_sizes, int n_in,
                              void* d_out, int out_size, void* d_ws, size_t ws_size,
                              hipStream_t stream)
{
    const float* x     = (const float*)d_in[0];
    const float* dwk   = (const float*)d_in[1];
    const float* dwb   = (const float*)d_in[2];
    const float* w0    = (const float*)d_in[3];
    const float* b0    = (const float*)d_in[4];
    const float* ws    = (const float*)d_in[5];
    const float* bs    = (const float*)d_in[6];
    const float* gamma = (const float*)d_in[7];
    const float* beta  = (const float*)d_in[8];
    const float* mmean = (const float*)d_in[9];
    const float* mvar  = (const float*)d_in[10];
    const float* wf    = (const float*)d_in[11];
    const float* bf    = (const float*)d_in[12];
    float* P   = (float*)d_ws;
    float* out = (float*)d_out;

    const int B = in_sizes[0] / (CA_H * CA_W);   // 16

    ca_fold<<<1, 32, 0, stream>>>(dwk, dwb, w0, b0, ws, bs,
                                  gamma, beta, mmean, mvar, wf, bf, P);

    dim3 grid(CA_H / CA_TH, B);
    const size_t shmem = (size_t)(CA_TH + 2) * CA_W * sizeof(float); // 73728 B
    ca_main<<<grid, 256, shmem, stream>>>(x, out, P);
}